// PreferenceAttentionLayer_19851338842169
// MI455X (gfx1250) — compile-verified
//
#include <hip/hip_runtime.h>

typedef __attribute__((ext_vector_type(2))) float v2f;
typedef __attribute__((ext_vector_type(8))) float v8f;

#define DDIM 1024
#define TLEN 2048
#define BATCH 16
#define MTOT (BATCH * TLEN)   // 32768 rows of x

// ---------------------------------------------------------------------------
// Kernel 1: qk projection.  qk[m, n] = x[m, :] . W[n, :] + b[n],  n in [0,2048)
// One wave computes a 16x64 output strip with 4 f32 WMMA accumulators.
// f32 16x16x4 operand layout: lane l holds matrix row (l&15), K elements
// {k + 2*(l>>4), k + 2*(l>>4) + 1} in VGPRs 0/1  -> one float2 load per step.
// ---------------------------------------------------------------------------
__global__ __launch_bounds__(256) void proj_qk_kernel(
    const float* __restrict__ x, const float* __restrict__ W,
    const float* __restrict__ bias, float* __restrict__ qk)
{
    const int wave   = (blockIdx.x * 256 + threadIdx.x) >> 5;
    const int lane   = threadIdx.x & 31;
    const int m_tile = wave >> 5;        // 2048 tiles of 16 rows
    const int n_grp  = wave & 31;        // 32 groups of 64 cols
    const int row    = lane & 15;
    const int kh     = (lane >> 4) << 1; // 0 or 2

    const float* arow  = x + (size_t)(m_tile * 16 + row) * DDIM + kh;
    const float* brow0 = W + (size_t)(n_grp * 64 + row) * DDIM + kh;

    v8f acc[4] = {};
    #pragma unroll 4
    for (int k = 0; k < DDIM; k += 4) {
        v2f a = *(const v2f*)(arow + k);
        #pragma unroll
        for (int j = 0; j < 4; ++j) {
            v2f bb = *(const v2f*)(brow0 + (size_t)j * 16 * DDIM + k);
            acc[j] = __builtin_amdgcn_wmma_f32_16x16x4_f32(
                false, a, false, bb, (short)0, acc[j], false, false);
        }
    }

    // C/D layout: VGPR r = row r (lanes 0-15) / row r+8 (lanes 16-31), col = lane&15
    const int mb = m_tile * 16 + ((lane >> 4) << 3);
    #pragma unroll
    for (int j = 0; j < 4; ++j) {
        const int n  = n_grp * 64 + j * 16 + row;
        const float bv = bias[n];
        #pragma unroll
        for (int r = 0; r < 8; ++r)
            qk[(size_t)(mb + r) * TLEN + n] = acc[j][r] + bv;
    }
}

// ---------------------------------------------------------------------------
// Kernel 2: v[m] = x[m, :] . W[2048, :] + b[2048]   (single projection column)
// One wave per row.
// ---------------------------------------------------------------------------
__global__ __launch_bounds__(256) void v_kernel(
    const float* __restrict__ x, const float* __restrict__ W,
    const float* __restrict__ bias, float* __restrict__ v)
{
    const int wave = (blockIdx.x * 256 + threadIdx.x) >> 5;
    const int lane = threadIdx.x & 31;
    const float* xr = x + (size_t)wave * DDIM;
    const float* wr = W + (size_t)2048 * DDIM;
    float s = 0.f;
    for (int i = lane; i < DDIM; i += 32) s += xr[i] * wr[i];
    #pragma unroll
    for (int m = 16; m >= 1; m >>= 1) s += __shfl_xor(s, m, 32);
    if (lane == 0) v[wave] = s + bias[2048];
}

// ---------------------------------------------------------------------------
// Kernel 3: attention.  One block = (batch b, 16-query tile).
// 8 waves fill S[16][2048] = Q K^T / 32 in LDS via f32 WMMA, then softmax
// per row (16 threads/row), stream normalized probs to attn, accumulate p.v.
// Block-level partial of sum_t(weighted) -> parts[block] (deterministic).
// ---------------------------------------------------------------------------
__global__ __launch_bounds__(256) void attn_kernel(
    const float* __restrict__ qk, const float* __restrict__ vvec,
    float* __restrict__ attn, float* __restrict__ parts)
{
    __shared__ float sc[16 * TLEN];    // 128 KB score tile (CDNA5: 320KB/WG ok)
    __shared__ float partial[16];

    const int b      = blockIdx.x >> 7;
    const int t_tile = blockIdx.x & 127;
    const int tid    = threadIdx.x;
    const int wave   = tid >> 5;
    const int lane   = tid & 31;
    const int row    = lane & 15;
    const int kh     = (lane >> 4) << 1;
    const int mb     = (lane >> 4) << 3;

    const float* Q = qk + (size_t)(b * TLEN + t_tile * 16 + row) * TLEN + kh;

    // ---- scores: each wave does 16 of the 128 s-tiles ----
    for (int st = wave; st < 128; st += 8) {
        const float* K = qk + (size_t)(b * TLEN + st * 16 + row) * TLEN + 1024 + kh;
        v8f c = {};
        #pragma unroll 4
        for (int k = 0; k < DDIM; k += 4) {
            v2f a  = *(const v2f*)(Q + k);
            v2f bb = *(const v2f*)(K + k);
            c = __builtin_amdgcn_wmma_f32_16x16x4_f32(
                false, a, false, bb, (short)0, c, false, false);
        }
        const int n = st * 16 + row;
        #pragma unroll
        for (int r = 0; r < 8; ++r)
            sc[(mb + r) * TLEN + n] = c[r] * 0.03125f;   // 1/sqrt(1024)
    }
    __syncthreads();

    // ---- softmax: 16 threads per row; groups of 16 lanes stay in one wave ----
    const int srow = tid >> 4;
    const int sub  = tid & 15;
    float* rp = sc + srow * TLEN;

    float mx = -3.0e38f;
    for (int i = sub; i < TLEN; i += 16) mx = fmaxf(mx, rp[i]);
    #pragma unroll
    for (int m = 8; m >= 1; m >>= 1) mx = fmaxf(mx, __shfl_xor(mx, m, 32));

    float sum = 0.f;
    for (int i = sub; i < TLEN; i += 16) {
        float e = __expf(rp[i] - mx);
        rp[i] = e;
        sum += e;
    }
    #pragma unroll
    for (int m = 8; m >= 1; m >>= 1) sum += __shfl_xor(sum, m, 32);
    const float inv = 1.f / sum;

    const float* vb = vvec + (size_t)b * TLEN;
    float* ao = attn + (size_t)(b * TLEN + t_tile * 16 + srow) * TLEN;
    float wsum = 0.f;
    for (int i = sub; i < TLEN; i += 16) {
        float p = rp[i] * inv;
        ao[i] = p;
        wsum += p * vb[i];
    }
    #pragma unroll
    for (int m = 8; m >= 1; m >>= 1) wsum += __shfl_xor(wsum, m, 32);
    if (sub == 0) partial[srow] = wsum;
    __syncthreads();

    if (tid == 0) {
        float s = 0.f;
        #pragma unroll
        for (int r = 0; r < 16; ++r) s += partial[r];
        parts[blockIdx.x] = s;      // sum over this block's 16 query rows
    }
}

// ---------------------------------------------------------------------------
// Kernel 4: reward[b] = (1/T) * sum over 128 block partials of batch b.
// Deterministic tree reduction, single block.
// ---------------------------------------------------------------------------
__global__ __launch_bounds__(256) void reward_kernel(
    const float* __restrict__ parts, float* __restrict__ reward)
{
    const int tid = threadIdx.x;
    const int b   = tid >> 4;
    const int sub = tid & 15;
    float s = 0.f;
    for (int i = sub; i < 128; i += 16) s += parts[b * 128 + i];
    #pragma unroll
    for (int m = 8; m >= 1; m >>= 1) s += __shfl_xor(s, m, 32);
    if (sub == 0) reward[b] = s * (1.0f / (float)TLEN);
}

// ---------------------------------------------------------------------------
extern "C" void kernel_launch(void* const* d_in, const int* in_sizes, int n_in,
                              void* d_out, int out_size, void* d_ws, size_t ws_size,
                              hipStream_t stream)
{
    const float* x    = (const float*)d_in[0];  // (16,2048,1024)
    const float* W    = (const float*)d_in[1];  // (2049,1024)
    const float* bias = (const float*)d_in[2];  // (2049,)

    float* reward = (float*)d_out;              // 16 floats
    float* attn   = reward + 16;                // 16*2048*2048 floats

    float* qk    = (float*)d_ws;                       // 32768*2048 f32 = 256 MB
    float* vv    = qk + (size_t)MTOT * TLEN;           // 32768 f32
    float* parts = vv + MTOT;                          // 2048 f32

    // 65536 waves (2048 m-tiles x 32 n-groups), 8 waves/block
    proj_qk_kernel<<<8192, 256, 0, stream>>>(x, W, bias, qk);
    // 32768 waves, 8 waves/block
    v_kernel<<<4096, 256, 0, stream>>>(x, W, bias, vv);
    // one block per (b, 16-query tile)
    attn_kernel<<<BATCH * (TLEN / 16), 256, 0, stream>>>(qk, vv, attn, parts);
    reward_kernel<<<1, 256, 0, stream>>>(parts, reward);
}